// MultiHeadAttention_66967130079453
// MI455X (gfx1250) — compile-verified
//
#include <hip/hip_runtime.h>
#include <hip/hip_bf16.h>

// ---------------------------------------------------------------------------
// MHA for MI455X (gfx1250), wave32, WMMA f32 16x16x4.
// B=4, L=2048, E=512, H=8, D=64.
// d_out = [out (B,L,E)] ++ [attn_weights (B,H,L,L)], both f32.
// HBM-bound by the 537MB attention-weights write -> NT stores for it.
// ---------------------------------------------------------------------------

typedef __attribute__((ext_vector_type(2))) float v2f;
typedef __attribute__((ext_vector_type(4))) float v4f;
typedef __attribute__((ext_vector_type(8))) float v8f;

#define L_SEQ   2048
#define E_DIM   512
#define H_NUM   8
#define D_HEAD  64
#define B_BATCH 4

#define XSTR 516   // LDS stride for 16x512 input tile (pad 4 -> conflict-free)
#define SSTR 2052  // LDS stride for 16x2048 score tile
#define QSTR 68    // LDS stride for 16x64 Q tile

__device__ __forceinline__ v8f wmma4(v2f a, v2f b, v8f c) {
  // D = A(16x4 f32) * B(4x16 f32) + C(16x16 f32)
  return __builtin_amdgcn_wmma_f32_16x16x4_f32(
      /*neg_a=*/false, a, /*neg_b=*/false, b,
      /*c_mod=*/(short)0, c, /*reuse_a=*/false, /*reuse_b=*/false);
}

// ---------------------------------------------------------------------------
// Projection GEMM: Y = X(8192x512) @ W(512x512) + bias.
// MODE 0: write Yrow[m*512 + e] row-major (final out; NT store, never re-read
//         by us).  X = ctx (just written) -> regular-temporal loads (L2 hit).
// MODE 1: write Yhead[b,h,l,d] head-split (q/k/v ws; re-read -> RT store).
//         X = raw q/k/v input, read exactly once -> NT loads.
// Block = 128 threads (4 waves); block owns one 16-row tile, each wave does
// 8 of the 32 16-col tiles. K-loop: 128 x v_wmma_f32_16x16x4_f32.
// ---------------------------------------------------------------------------
template <int MODE>
__global__ __launch_bounds__(128) void mha_proj_kernel(
    const float* __restrict__ X, const float* __restrict__ W,
    const float* __restrict__ bias, float* __restrict__ Y) {
  __shared__ float sX[16 * XSTR];

  const int tid  = threadIdx.x;
  const int lane = tid & 31;
  const int wave = tid >> 5;
  const int half = lane >> 4;   // 0: K={0,1}, 1: K={2,3}
  const int l16  = lane & 15;
  const int m0   = blockIdx.x * 16;

  // Stage 16x512 tile of X into LDS (v4f, coalesced).
  const v4f* src = (const v4f*)(X + (size_t)m0 * E_DIM);
  #pragma unroll
  for (int i = 0; i < 16; ++i) {
    int j   = tid + i * 128;     // v4f index 0..2047
    int row = j >> 7;
    int c4  = j & 127;
    v4f v = (MODE == 1) ? __builtin_nontemporal_load(&src[row * 128 + c4])
                        : src[row * 128 + c4];
    *(v4f*)&sX[row * XSTR + c4 * 4] = v;
  }
  __syncthreads();

  for (int nt = wave; nt < 32; nt += 4) {
    const int n0 = nt * 16;
    __builtin_prefetch(W + n0 + l16, 0, 1);   // global_prefetch_b8

    float bval = bias[n0 + l16];
    v8f c;
    #pragma unroll
    for (int r = 0; r < 8; ++r) c[r] = bval;  // bias broadcast across rows

    const float* wp = W + n0 + l16;           // this lane's output column
    for (int k0 = 0; k0 < E_DIM; k0 += 4) {
      const int ksel = k0 + half * 2;
      v2f a = *(const v2f*)&sX[l16 * XSTR + ksel];     // A: 16x4 from LDS
      v2f b;
      b.x = wp[(size_t)ksel * E_DIM];                  // B: 4x16 from W
      b.y = wp[(size_t)(ksel + 1) * E_DIM];
      c = wmma4(a, b, c);
    }

    #pragma unroll
    for (int r = 0; r < 8; ++r) {
      const int row = r + half * 8;
      const int m = m0 + row;
      const int e = n0 + l16;
      if (MODE == 0) {
        // final output: written once, never re-read -> non-temporal
        __builtin_nontemporal_store(c[r], &Y[(size_t)m * E_DIM + e]);
      } else {
        const int b_ = m >> 11, l_ = m & (L_SEQ - 1);
        const int h_ = e >> 6,  d_ = e & (D_HEAD - 1);
        Y[(((size_t)b_ * H_NUM + h_) * L_SEQ + l_) * D_HEAD + d_] = c[r];
      }
    }
  }
}

// ---------------------------------------------------------------------------
// Attention: per block = one (b,h) and one 16-row q tile over all 2048 keys.
//   scores = Q Kt * 0.125 + gene, mask==0 -> -BIG, softmax, weights -> d_out,
//   ctx = P @ V -> workspace [B,L,E].
// Dynamic LDS: 16x2052 scores + 16x68 Q + reduction scratch = 136320 B
// (2 blocks / 320KB WGP).
// ---------------------------------------------------------------------------
__global__ __launch_bounds__(128) void mha_attn_kernel(
    const float* __restrict__ q_ws, const float* __restrict__ k_ws,
    const float* __restrict__ v_ws, const int* __restrict__ mask,
    const float* __restrict__ gene, float* __restrict__ wout,
    float* __restrict__ ctx) {
  extern __shared__ float smem[];
  float* sS   = smem;                  // 16*2052
  float* sQ   = sS + 16 * SSTR;        // 16*68
  float* red  = sQ + 16 * QSTR;        // 16*8 partials
  float* rmax = red + 128;             // 16
  float* rinv = rmax + 16;             // 16

  const int tid  = threadIdx.x;
  const int lane = tid & 31;
  const int wave = tid >> 5;
  const int half = lane >> 4;
  const int l16  = lane & 15;

  const int qt = blockIdx.x;           // q tile 0..127
  const int bh = blockIdx.y;           // 0..31
  const int b  = bh >> 3;
  const int h  = bh & 7;
  const size_t bhL = (size_t)bh * L_SEQ;
  const int q0 = qt * 16;

  // Stage Q tile (16x64) into LDS.
  {
    const v4f* qsrc = (const v4f*)(q_ws + (bhL + q0) * D_HEAD);
    #pragma unroll
    for (int i = 0; i < 2; ++i) {
      int j = tid + i * 128;           // v4f idx 0..255
      int row = j >> 4, c4 = j & 15;
      *(v4f*)&sQ[row * QSTR + c4 * 4] = qsrc[row * 16 + c4];
    }
  }
  __syncthreads();

  // ---- Scores: each wave owns 32 contiguous 16-wide key tiles ----
  const float scale = 0.125f;          // 1/sqrt(64)
  for (int kt = wave * 32; kt < wave * 32 + 32; ++kt) {
    v8f c;
    #pragma unroll
    for (int r = 0; r < 8; ++r) c[r] = 0.0f;

    // B operand: Kt(d, n) = K[kt*16+n][d]; contiguous v2f in [B,H,L,D].
    const float* kbase = k_ws + (bhL + kt * 16 + l16) * D_HEAD + half * 2;
    #pragma unroll 4
    for (int d0 = 0; d0 < D_HEAD; d0 += 4) {
      v2f a = *(const v2f*)&sQ[l16 * QSTR + d0 + half * 2];
      v2f bb = *(const v2f*)&kbase[d0];
      c = wmma4(a, bb, c);
    }

    const int kc = kt * 16 + l16;
    #pragma unroll
    for (int r = 0; r < 8; ++r) {
      const int row = r + half * 8;
      const int qr  = q0 + row;
      float s = c[r] * scale + gene[(size_t)qr * L_SEQ + kc];
      int mval = mask[((size_t)b * L_SEQ + qr) * L_SEQ + kc];
      s = (mval == 0) ? -3.0e38f : s;
      sS[row * SSTR + kc] = s;
    }
  }
  __syncthreads();

  // ---- Softmax over each of 16 rows (8 threads/row, 256 cols each) ----
  const int srow = tid & 15;
  const int g    = tid >> 4;           // 0..7
  float* rowp = &sS[srow * SSTR];

  float mx = -3.4e38f;
  for (int i = 0; i < 256; ++i) mx = fmaxf(mx, rowp[g * 256 + i]);
  red[srow * 8 + g] = mx;
  __syncthreads();
  if (tid < 16) {
    float m2 = red[tid * 8];
    #pragma unroll
    for (int j = 1; j < 8; ++j) m2 = fmaxf(m2, red[tid * 8 + j]);
    rmax[tid] = m2;
  }
  __syncthreads();

  const float mrow = rmax[srow];
  float sum = 0.0f;
  for (int i = 0; i < 256; ++i) {
    float e = __expf(rowp[g * 256 + i] - mrow);
    rowp[g * 256 + i] = e;
    sum += e;
  }
  red[srow * 8 + g] = sum;
  __syncthreads();
  if (tid < 16) {
    float s2 = 0.0f;
    #pragma unroll
    for (int j = 0; j < 8; ++j) s2 += red[tid * 8 + j];
    rinv[tid] = (s2 > 0.0f) ? (1.0f / s2) : 0.0f;
  }
  __syncthreads();

  // ---- Normalize in LDS + stream weights tile to d_out ----
  // 537MB total, written once, never re-read -> non-temporal v4f stores
  // so gene/mask/K/V stay resident in the 192MB L2.
  const float inv = rinv[srow];
  v4f* wdst = (v4f*)(wout + ((bhL + q0 + srow) * L_SEQ) + g * 256);
  for (int i = 0; i < 64; ++i) {
    v4f p = *(v4f*)&rowp[g * 256 + i * 4];
    p = p * inv;
    *(v4f*)&rowp[g * 256 + i * 4] = p;
    __builtin_nontemporal_store(p, &wdst[i]);
  }
  __syncthreads();

  // ---- ctx = P @ V : each wave owns one 16x16 slice of the 16x64 output ----
  const int d0 = wave * 16;
  v8f c;
  #pragma unroll
  for (int r = 0; r < 8; ++r) c[r] = 0.0f;

  const float* vbase = v_ws + bhL * D_HEAD + d0 + l16;
  for (int k0 = 0; k0 < L_SEQ; k0 += 4) {
    v2f a = *(const v2f*)&sS[l16 * SSTR + k0 + half * 2];  // P fragment
    const size_t off = (size_t)(k0 + half * 2) * D_HEAD;
    v2f bb;
    bb.x = vbase[off];
    bb.y = vbase[off + D_HEAD];
    c = wmma4(a, bb, c);
  }

  #pragma unroll
  for (int r = 0; r < 8; ++r) {
    const int row = r + half * 8;
    ctx[((size_t)b * L_SEQ + q0 + row) * E_DIM + h * D_HEAD + d0 + l16] = c[r];
  }
}

// ---------------------------------------------------------------------------
extern "C" void kernel_launch(void* const* d_in, const int* in_sizes, int n_in,
                              void* d_out, int out_size, void* d_ws,
                              size_t ws_size, hipStream_t stream) {
  const float* query = (const float*)d_in[0];
  const float* key   = (const float*)d_in[1];
  const float* value = (const float*)d_in[2];
  const int*   amask = (const int*)d_in[3];
  const float* gene  = (const float*)d_in[4];
  const float* Wq = (const float*)d_in[5];
  const float* bq = (const float*)d_in[6];
  const float* Wk = (const float*)d_in[7];
  const float* bk = (const float*)d_in[8];
  const float* Wv = (const float*)d_in[9];
  const float* bv = (const float*)d_in[10];
  const float* Wo = (const float*)d_in[11];
  const float* bo = (const float*)d_in[12];

  float* out  = (float*)d_out;                                  // (B,L,E)
  float* wout = out + (size_t)B_BATCH * L_SEQ * E_DIM;          // (B,H,L,L)

  const size_t QKV = (size_t)B_BATCH * L_SEQ * E_DIM;           // 4194304
  float* ws   = (float*)d_ws;
  float* q_ws = ws;
  float* k_ws = ws + QKV;
  float* v_ws = ws + 2 * QKV;
  float* ctx  = ws + 3 * QKV;

  dim3 blk(128);
  dim3 pgrid((B_BATCH * L_SEQ) / 16);  // 512 row-tiles

  mha_proj_kernel<1><<<pgrid, blk, 0, stream>>>(query, Wq, bq, q_ws);
  mha_proj_kernel<1><<<pgrid, blk, 0, stream>>>(key,   Wk, bk, k_ws);
  mha_proj_kernel<1><<<pgrid, blk, 0, stream>>>(value, Wv, bv, v_ws);

  const size_t smem = (size_t)(16 * SSTR + 16 * QSTR + 128 + 32) * sizeof(float);
  mha_attn_kernel<<<dim3(L_SEQ / 16, B_BATCH * H_NUM), blk, smem, stream>>>(
      q_ws, k_ws, v_ws, amask, gene, wout, ctx);

  mha_proj_kernel<0><<<pgrid, blk, 0, stream>>>(ctx, Wo, bo, out);
}